// LovaszLoss_38697655336983
// MI455X (gfx1250) — compile-verified
//
#include <hip/hip_runtime.h>
#include <stdint.h>

// Problem constants (from reference): B=2, C=19, F=4, H=256, W=384
#define N_PIX    786432            // B*F*H*W
#define NCLASS   19
#define FHW      393216            // F*H*W (channel stride)
#define BSTRIDE  (NCLASS*FHW)      // batch stride in input
#define NB256    3072              // N_PIX / 256

// Radix sort config: 768 blocks x (256 thr x 4 items) = 786432 per class
#define SORT_BLOCKS 768
#define SORT_ITEMS  4
#define SORT_CHUNK  1024
#define HISTSZ      (256 * SORT_BLOCKS)

typedef float v2f __attribute__((ext_vector_type(2)));
typedef float v8f __attribute__((ext_vector_type(8)));

// ---------------------------------------------------------------- utilities
__global__ void zero_small(unsigned* gts) {
  int t = threadIdx.x;
  if (t < NCLASS) gts[t] = 0u;
}

// per-class foreground counts (gts[c] = #pixels with label c)
__global__ void class_hist(const int* __restrict__ tgt, unsigned* __restrict__ gts) {
  __shared__ unsigned h[NCLASS];
  int t = threadIdx.x;
  if (t < NCLASS) h[t] = 0u;
  __syncthreads();
  int n = blockIdx.x * 256 + t;
  atomicAdd(&h[tgt[n]], 1u);
  __syncthreads();
  if (t < NCLASS) atomicAdd(&gts[t], h[t]);
}

// Build 64-bit records: [~float_bits(error) : 32 | fg : 1]
// Ascending sort on top 32 bits == descending sort on error (error >= 0).
// blockIdx.y = class slab.
__global__ void build_keys(const float* __restrict__ inp, const int* __restrict__ tgt,
                           unsigned long long* __restrict__ rec, int cbase) {
  int c = cbase + blockIdx.y;
  unsigned long long* r = rec + (size_t)blockIdx.y * N_PIX;
  int n = blockIdx.x * 256 + threadIdx.x;
  int b = n / FHW;
  int q = n - b * FHW;
  float p = inp[(size_t)b * BSTRIDE + (size_t)c * FHW + (size_t)q];
  unsigned fg = (tgt[n] == c) ? 1u : 0u;
  float e = fabsf((float)fg - p);
  unsigned k = ~__float_as_uint(e);
  r[n] = ((unsigned long long)k << 32) | (unsigned long long)fg;
}

// --------------------------------------------------------------- radix sort
__global__ void radix_count(const unsigned long long* __restrict__ recAll,
                            unsigned* __restrict__ histAll, int shift) {
  const unsigned long long* rec = recAll + (size_t)blockIdx.y * N_PIX;
  unsigned* hist = histAll + (size_t)blockIdx.y * HISTSZ;
  __shared__ unsigned lcnt[256];
  int t = threadIdx.x;
  lcnt[t] = 0u;
  __syncthreads();
  size_t base = (size_t)blockIdx.x * SORT_CHUNK;
  for (int r = 0; r < SORT_ITEMS; ++r) {
    unsigned long long v = rec[base + (size_t)r * 256 + t];
    unsigned d = (unsigned)(v >> shift) & 0xFFu;
    atomicAdd(&lcnt[d], 1u);
  }
  __syncthreads();
  hist[(size_t)t * SORT_BLOCKS + blockIdx.x] = lcnt[t];  // bin-major
}

// one block per class slab: exclusive prefix over hist in (bin, block) order
__global__ void radix_scan(unsigned* __restrict__ histAll) {
  unsigned* hist = histAll + (size_t)blockIdx.x * HISTSZ;
  __shared__ unsigned tot[256];
  int t = threadIdx.x;
  unsigned s = 0u;
  for (int b = 0; b < SORT_BLOCKS; ++b) s += hist[(size_t)t * SORT_BLOCKS + b];
  tot[t] = s;
  __syncthreads();
  for (int off = 1; off < 256; off <<= 1) {   // Hillis-Steele inclusive
    unsigned v = (t >= off) ? tot[t - off] : 0u;
    __syncthreads();
    tot[t] += v;
    __syncthreads();
  }
  unsigned run = (t == 0) ? 0u : tot[t - 1];
  for (int b = 0; b < SORT_BLOCKS; ++b) {
    unsigned v = hist[(size_t)t * SORT_BLOCKS + b];
    hist[(size_t)t * SORT_BLOCKS + b] = run;
    run += v;
  }
}

// Stable scatter: ranks via wave32 ballot digit-matching + per-wave LDS counts.
__global__ void radix_scatter(const unsigned long long* __restrict__ inAll,
                              unsigned long long* __restrict__ outAll,
                              const unsigned* __restrict__ histAll, int shift) {
  const unsigned long long* in = inAll + (size_t)blockIdx.y * N_PIX;
  unsigned long long* out = outAll + (size_t)blockIdx.y * N_PIX;
  const unsigned* hist = histAll + (size_t)blockIdx.y * HISTSZ;
  __shared__ unsigned base[256];
  __shared__ unsigned run[256];
  __shared__ unsigned wcnt[8 * 256];
  int t = threadIdx.x;
  int lane = t & 31;
  int wave = t >> 5;
  base[t] = hist[(size_t)t * SORT_BLOCKS + blockIdx.x];
  run[t] = 0u;
  __syncthreads();
  size_t cbase = (size_t)blockIdx.x * SORT_CHUNK;
  unsigned ltmask = (lane == 0) ? 0u : (0xFFFFFFFFu >> (32 - lane));
  for (int r = 0; r < SORT_ITEMS; ++r) {
    unsigned long long v = in[cbase + (size_t)r * 256 + t];
    if (r + 1 < SORT_ITEMS)                         // global_prefetch_b8
      __builtin_prefetch(&in[cbase + (size_t)(r + 1) * 256 + t], 0, 3);
    unsigned d = (unsigned)(v >> shift) & 0xFFu;
    for (int i = 0; i < 8; ++i) wcnt[i * 256 + t] = 0u;
    __syncthreads();
    // mask of lanes in this wave holding the same digit (8 wave32 ballots)
    unsigned mask = 0xFFFFFFFFu;
    for (int bit = 0; bit < 8; ++bit) {
      unsigned vote = (unsigned)__ballot((d >> bit) & 1u);
      mask &= ((d >> bit) & 1u) ? vote : ~vote;
    }
    unsigned laneRank = __popc(mask & ltmask);
    if (laneRank == 0) wcnt[wave * 256 + d] = __popc(mask);
    __syncthreads();
    unsigned prior = 0u;
    for (int w = 0; w < wave; ++w) prior += wcnt[w * 256 + d];
    out[base[d] + run[d] + prior + laneRank] = v;
    __syncthreads();
    unsigned add = 0u;
    for (int w = 0; w < 8; ++w) add += wcnt[w * 256 + t];
    run[t] += add;
    __syncthreads();
  }
}

// ------------------------------------------------ fg prefix scan (2 levels)
__global__ void fg_partials(const unsigned long long* __restrict__ recAll,
                            unsigned* __restrict__ blkAll) {
  const unsigned long long* rec = recAll + (size_t)blockIdx.y * N_PIX;
  unsigned* blk = blkAll + (size_t)blockIdx.y * NB256;
  __shared__ unsigned wtot[8];
  int t = threadIdx.x, lane = t & 31, wave = t >> 5;
  unsigned fg = (unsigned)(rec[blockIdx.x * 256 + t] & 1ull);
  unsigned m = (unsigned)__ballot(fg);
  if (lane == 0) wtot[wave] = __popc(m);
  __syncthreads();
  if (t == 0) {
    unsigned s = 0u;
    for (int w = 0; w < 8; ++w) s += wtot[w];
    blk[blockIdx.x] = s;
  }
}

__global__ void fg_scan(const unsigned* __restrict__ inAll, unsigned* __restrict__ outAll) {
  const unsigned* in = inAll + (size_t)blockIdx.x * NB256;
  unsigned* out = outAll + (size_t)blockIdx.x * NB256;
  __shared__ unsigned st[256];
  int t = threadIdx.x;
  unsigned s = 0u;
  for (int i = 0; i < 12; ++i) s += in[t * 12 + i];   // 256*12 = 3072
  st[t] = s;
  __syncthreads();
  for (int off = 1; off < 256; off <<= 1) {
    unsigned v = (t >= off) ? st[t - off] : 0u;
    __syncthreads();
    st[t] += v;
    __syncthreads();
  }
  unsigned run = (t == 0) ? 0u : st[t - 1];
  for (int i = 0; i < 12; ++i) { unsigned v = in[t * 12 + i]; out[t * 12 + i] = run; run += v; }
}

// ------------------------------- per-element Lovasz gradient + WMMA reduce
// contribution_i = e_i * (J(i, F_i) - J(i-1, F_i - fg_i)),
// J(n, F) = 1 - (p - F) / (p + n - F)  (telescopes to the reference sum).
__global__ void contrib_kernel(const unsigned long long* __restrict__ recAll,
                               const unsigned* __restrict__ exAll,
                               const unsigned* __restrict__ gts,
                               float* __restrict__ bsumAll, int cb) {
  const unsigned long long* rec = recAll + (size_t)blockIdx.y * N_PIX;
  const unsigned* ex = exAll + (size_t)blockIdx.y * NB256;
  float* bsum = bsumAll + (size_t)blockIdx.y * NB256;
  int c = cb + blockIdx.y;
  __shared__ unsigned wtot[8];
  __shared__ float wsum[8];
  int t = threadIdx.x, lane = t & 31, wave = t >> 5;
  int i = blockIdx.x * 256 + t;                  // sorted position (0-indexed)
  unsigned long long v = rec[i];
  unsigned fg = (unsigned)(v & 1ull);
  float e = __uint_as_float(~(unsigned)(v >> 32));
  // inclusive fg prefix within wave/block via wave32 ballot
  unsigned m = (unsigned)__ballot(fg);
  unsigned lemask = 0xFFFFFFFFu >> (31 - lane);
  unsigned incl = __popc(m & lemask);
  if (lane == 0) wtot[wave] = __popc(m);
  __syncthreads();
  unsigned prior = 0u;
  for (int w = 0; w < wave; ++w) prior += wtot[w];
  float F  = (float)(ex[blockIdx.x] + prior + incl);
  float p  = (float)gts[c];
  float F0 = F - (float)fg;
  float u1 = fmaxf(p + (float)(i + 1) - F,  1e-12f);
  float u0 = fmaxf(p + (float)i       - F0, 1e-12f);
  float j1 = 1.0f - (p - F)  / u1;
  float j0 = 1.0f - (p - F0) / u0;
  float contrib = (p > 0.0f) ? e * (j1 - j0) : 0.0f;  // absent class masked

  // Wave32 reduction via V_WMMA_F32_16X16X4_F32: A[m,{0,2}] = lane values,
  // B = ones => D[m,*] = val(m) + val(m+16); rows summed + shfl_xor(16).
  v2f a;  a.x = contrib; a.y = 0.0f;
  v2f bm; bm.x = 1.0f;   bm.y = 1.0f;
  v8f cz = {};
  v8f dmat = __builtin_amdgcn_wmma_f32_16x16x4_f32(false, a, false, bm,
                                                   (short)0, cz, false, false);
  float rt = dmat[0] + dmat[1] + dmat[2] + dmat[3]
           + dmat[4] + dmat[5] + dmat[6] + dmat[7];
  rt += __shfl_xor(rt, 16, 32);                  // full 32-lane total
  if (lane == 0) wsum[wave] = rt;
  __syncthreads();
  if (t == 0) {
    float s = 0.0f;
    for (int w = 0; w < 8; ++w) s += wsum[w];
    bsum[blockIdx.x] = s;                        // deterministic (no atomics)
  }
}

__global__ void reduce_cls(const float* __restrict__ bsAll, float* __restrict__ cls, int cb) {
  const float* bs = bsAll + (size_t)blockIdx.x * NB256;
  int c = cb + blockIdx.x;
  __shared__ float st[256];
  int t = threadIdx.x;
  float s = 0.0f;
  for (int i = 0; i < 12; ++i) s += bs[t * 12 + i];
  st[t] = s;
  __syncthreads();
  for (int off = 128; off > 0; off >>= 1) {
    if (t < off) st[t] += st[t + off];
    __syncthreads();
  }
  if (t == 0) cls[c] = st[0];
}

__global__ void finalize(const unsigned* __restrict__ gts,
                         const float* __restrict__ cls, float* __restrict__ out) {
  if (threadIdx.x == 0) {
    float s = 0.0f, np = 0.0f;
    for (int c = 0; c < NCLASS; ++c)
      if (gts[c] > 0u) { s += cls[c]; np += 1.0f; }
    out[0] = s / fmaxf(np, 1.0f);
  }
}

// ------------------------------------------------------------------- driver
extern "C" void kernel_launch(void* const* d_in, const int* in_sizes, int n_in,
                              void* d_out, int out_size, void* d_ws, size_t ws_size,
                              hipStream_t stream) {
  const float* inp = (const float*)d_in[0];
  const int*   tgt = (const int*)d_in[1];

  // Batch all 19 independent per-class sorts if workspace allows (one grid,
  // ~19 launches total); otherwise process classes sequentially (~300).
  size_t need19 = (size_t)19 * N_PIX * 8 * 2     // record ping-pong
                + (size_t)19 * HISTSZ * 4        // radix histograms
                + (size_t)19 * NB256 * 4 * 3     // blkFg, blkEx, bsum
                + 1024;                          // gts, cls, slack
  int NS = (ws_size >= need19) ? 19 : 1;

  char* w = (char*)d_ws;
  unsigned long long* bufA = (unsigned long long*)w; w += (size_t)NS * N_PIX * 8;
  unsigned long long* bufB = (unsigned long long*)w; w += (size_t)NS * N_PIX * 8;
  unsigned* hist  = (unsigned*)w; w += (size_t)NS * HISTSZ * 4;
  unsigned* blkFg = (unsigned*)w; w += (size_t)NS * NB256 * 4;
  unsigned* blkEx = (unsigned*)w; w += (size_t)NS * NB256 * 4;
  float*    bsum  = (float*)w;    w += (size_t)NS * NB256 * 4;
  unsigned* gts   = (unsigned*)w; w += 64 * 4;
  float*    cls   = (float*)w;    w += 64 * 4;

  zero_small<<<1, 32, 0, stream>>>(gts);
  class_hist<<<NB256, 256, 0, stream>>>(tgt, gts);

  for (int cb = 0; cb < NCLASS; cb += NS) {
    int ns = (NCLASS - cb < NS) ? (NCLASS - cb) : NS;
    dim3 gPix(NB256, ns);
    dim3 gSort(SORT_BLOCKS, ns);

    build_keys<<<gPix, 256, 0, stream>>>(inp, tgt, bufA, cb);
    unsigned long long* cur = bufA;
    unsigned long long* alt = bufB;
    for (int pass = 0; pass < 4; ++pass) {
      int shift = 32 + 8 * pass;
      radix_count<<<gSort, 256, 0, stream>>>(cur, hist, shift);
      radix_scan<<<ns, 256, 0, stream>>>(hist);
      radix_scatter<<<gSort, 256, 0, stream>>>(cur, alt, hist, shift);
      unsigned long long* tmp = cur; cur = alt; alt = tmp;
    }
    fg_partials<<<gPix, 256, 0, stream>>>(cur, blkFg);
    fg_scan<<<ns, 256, 0, stream>>>(blkFg, blkEx);
    contrib_kernel<<<gPix, 256, 0, stream>>>(cur, blkEx, gts, bsum, cb);
    reduce_cls<<<ns, 256, 0, stream>>>(bsum, cls, cb);
  }
  finalize<<<1, 32, 0, stream>>>(gts, cls, (float*)d_out);
}